// AdaProp_59485297050019
// MI455X (gfx1250) — compile-verified
//
#include <hip/hip_runtime.h>
#include <hip/hip_bf16.h>
#include <math.h>

typedef __attribute__((ext_vector_type(2))) float v2f;
typedef __attribute__((ext_vector_type(8))) float v8f;

#define DA 128   // D == A == 128 in this problem
#define TN 8     // 16-col tiles per 16-row strip (covers all 128 cols)

// ---------------------------------------------------------------------------
// Zero-fill (agg buffer must be zeroed every call; harness poisons d_ws).
// ---------------------------------------------------------------------------
__global__ void zero_f32(float* __restrict__ p, long n) {
    long i = (long)blockIdx.x * blockDim.x + threadIdx.x;
    long stride = (long)gridDim.x * blockDim.x;
    for (; i < n; i += stride) p[i] = 0.0f;
}

// ---------------------------------------------------------------------------
// Small projections:
//   rproj[r,:]  = rela_embed[r,:] @ Wr                      (R rows)
//   qproj[b,:]  = rela_embed[q_rel[b],:] @ Wqr_w + Wqr_b    (B rows)
// ---------------------------------------------------------------------------
__global__ void small_proj(const float* __restrict__ rela,
                           const float* __restrict__ Wr,
                           const int*   __restrict__ q_rel,
                           const float* __restrict__ Wqr_w,
                           const float* __restrict__ Wqr_b,
                           float* __restrict__ rproj,
                           float* __restrict__ qproj,
                           int R) {
    int row = blockIdx.x;
    int n   = threadIdx.x;              // 0..127
    const float* src;
    const float* W;
    float*       dst;
    float        acc;
    if (row < R) {
        src = rela + (size_t)row * DA;
        W   = Wr;
        dst = rproj + (size_t)row * DA;
        acc = 0.0f;
    } else {
        int b = row - R;
        int q = q_rel[b];
        src = rela + (size_t)q * DA;
        W   = Wqr_w;
        dst = qproj + (size_t)b * DA;
        acc = Wqr_b[n];
    }
    #pragma unroll 8
    for (int k = 0; k < DA; ++k)
        acc = fmaf(src[k], W[(size_t)k * DA + n], acc);
    dst[n] = acc;
}

// ---------------------------------------------------------------------------
// Row-major fp32 GEMM  Y[M x 128] = X[M x 128] @ W[128 x 128]  (opt. ReLU)
// via V_WMMA_F32_16X16X4_F32.
//
//  * W is staged once per workgroup into LDS in a k-pair-interleaved layout:
//      element (k, n)  ->  dword ((k>>1)*128 + n)*2 + (k&1)
//    so each lane's B fragment (W[k][n], W[k+1][n]) is one aligned
//    ds_load_b64 at a compile-time constant offset (max < 64 KB).
//  * One wave32 computes a full 16-row x 128-col strip: the A fragment is
//    loaded once per k-step (global_load_b64) and reused by 8 WMMAs
//    (8 v8f accumulators resident).
//
// Fragment layout (ISA 7.12.2):
//   A 16x4 : lane l -> row M = l&15 ; VGPR0 = K = 2*(l>>4), VGPR1 = K+1
//   B 4x16 : lane l -> col N = l&15 ; VGPR0 = K = 2*(l>>4), VGPR1 = K+1
//   C 16x16: lane l, VGPR v -> row v + 8*(l>>4), col l&15
// M must be a multiple of 16 (true here: 50000).
// ---------------------------------------------------------------------------
__global__ __launch_bounds__(256) void
gemm128_wmma(const float* __restrict__ X,
             const float* __restrict__ W,
             float* __restrict__ Y,
             int M, int do_relu) {
    __shared__ float Wlds[DA * DA];          // 64 KB of the WGP's 320 KB

    const int tid = threadIdx.x;

    // Cooperative stage of W into pair-interleaved LDS layout.
    // pair p = k>>1: read W[2p][n], W[2p+1][n], store contiguously (b64).
    for (int idx = tid; idx < (DA / 2) * DA; idx += 256) {
        const int p = idx >> 7;              // k-pair index (0..63)
        const int n = idx & (DA - 1);        // column
        float2 w2;
        w2.x = W[(size_t)(2 * p + 0) * DA + n];
        w2.y = W[(size_t)(2 * p + 1) * DA + n];
        *(float2*)(Wlds + ((size_t)p * DA + n) * 2) = w2;
    }
    __syncthreads();

    const int strip = blockIdx.x * 8 + (tid >> 5);   // 16-row strip index
    if (strip * 16 >= M) return;                     // wave-uniform guard
    const int lane = tid & 31;
    const int m0   = strip * 16;
    const int half = lane >> 4;                      // 0 or 1
    const int r    = lane & 15;                      // row (A) / col (B,C)

    v8f c[TN];
    v8f zero = {};
    #pragma unroll
    for (int t = 0; t < TN; ++t) c[t] = zero;

    const float* xrow = X + (size_t)(m0 + r) * DA + 2 * half;  // global (b64/step)
    // lane-dependent base in v2f units; all inner offsets are constants
    const v2f* wl = ((const v2f*)Wlds) + half * DA + r;

    #pragma unroll 4
    for (int k0 = 0; k0 < DA; k0 += 4) {
        v2f a;
        a.x = xrow[k0 + 0];
        a.y = xrow[k0 + 1];
        #pragma unroll
        for (int t = 0; t < TN; ++t) {
            v2f b = wl[(k0 >> 1) * DA + t * 16];   // ds_load_b64, const offset
            c[t] = __builtin_amdgcn_wmma_f32_16x16x4_f32(
                       /*neg_a=*/false, a, /*neg_b=*/false, b,
                       /*c_mod=*/(short)0, c[t],
                       /*reuse_a=*/false, /*reuse_b=*/false);
        }
    }

    #pragma unroll
    for (int t = 0; t < TN; ++t) {
        #pragma unroll
        for (int v = 0; v < 8; ++v) {
            float val = c[t][v];
            if (do_relu) val = fmaxf(val, 0.0f);
            Y[(size_t)(m0 + v + 8 * half) * DA + t * 16 + r] = val;
        }
    }
}

// ---------------------------------------------------------------------------
// Per-edge fused kernel. One wave32 per edge (4 floats per lane = 128):
//   pre   = relu(sproj[sub] + rproj[rel] + qproj[r_idx])
//   alpha = sigmoid(dot(pre, Wa))        (wave shfl-xor reduction)
//   agg[obj] += (hidden[sub] + rela[rel]) * alpha   (global f32 atomics -> L2)
// qproj (B x 128, 32 KB) is staged in dynamic LDS (hot for every edge).
// edges row e: [r_idx, _, rel, _, sub, obj]
// ---------------------------------------------------------------------------
__global__ __launch_bounds__(256) void
edge_msg(const int*   __restrict__ edges,
         const float* __restrict__ sproj,
         const float* __restrict__ rproj,
         const float* __restrict__ qproj,
         const float* __restrict__ hidden,
         const float* __restrict__ rela,
         const float* __restrict__ Wa,
         float* __restrict__ agg,
         long E, int B) {
    extern __shared__ float qlds[];              // B * 128 floats

    const int tid = threadIdx.x;
    for (int i = tid; i < B * DA; i += 256)
        qlds[i] = qproj[i];
    __syncthreads();

    const int  lane   = tid & 31;
    const long wave   = (long)blockIdx.x * (blockDim.x >> 5) + (tid >> 5);
    const long nwaves = (long)gridDim.x * (blockDim.x >> 5);
    const int  j      = lane * 4;

    const float4 wa = *(const float4*)(Wa + j);  // per-lane constant slice

    for (long e = wave; e < E; e += nwaves) {
        const int* ed  = edges + e * 6;
        const int ridx = ed[0];
        const int rel  = ed[2];
        const int sub  = ed[4];
        const int obj  = ed[5];

        const float4 sp = *(const float4*)(sproj + (size_t)sub * DA + j);
        const float4 rp = *(const float4*)(rproj + (size_t)rel * DA + j);
        const float4 qp = *(const float4*)(qlds  + ridx * DA + j);   // LDS

        const float p0 = fmaxf(sp.x + rp.x + qp.x, 0.0f);
        const float p1 = fmaxf(sp.y + rp.y + qp.y, 0.0f);
        const float p2 = fmaxf(sp.z + rp.z + qp.z, 0.0f);
        const float p3 = fmaxf(sp.w + rp.w + qp.w, 0.0f);

        float part = p0 * wa.x + p1 * wa.y + p2 * wa.z + p3 * wa.w;
        #pragma unroll
        for (int off = 16; off; off >>= 1)
            part += __shfl_xor(part, off, 32);
        const float alpha = 1.0f / (1.0f + __expf(-part));

        const float4 h  = *(const float4*)(hidden + (size_t)sub * DA + j);
        const float4 hr = *(const float4*)(rela   + (size_t)rel * DA + j);

        float* dst = agg + (size_t)obj * DA + j;
        atomicAdd(dst + 0, (h.x + hr.x) * alpha);
        atomicAdd(dst + 1, (h.y + hr.y) * alpha);
        atomicAdd(dst + 2, (h.z + hr.z) * alpha);
        atomicAdd(dst + 3, (h.w + hr.w) * alpha);
    }
}

// ---------------------------------------------------------------------------
// Launch
// ---------------------------------------------------------------------------
extern "C" void kernel_launch(void* const* d_in, const int* in_sizes, int n_in,
                              void* d_out, int out_size, void* d_ws, size_t ws_size,
                              hipStream_t stream) {
    const int*   q_rel  = (const int*)  d_in[0];
    const float* hidden = (const float*)d_in[1];
    const int*   edges  = (const int*)  d_in[2];
    const float* rela   = (const float*)d_in[3];
    const float* Ws     = (const float*)d_in[4];
    const float* Wr     = (const float*)d_in[5];
    const float* Wqr_w  = (const float*)d_in[6];
    const float* Wqr_b  = (const float*)d_in[7];
    const float* Wa     = (const float*)d_in[8];
    const float* Wh     = (const float*)d_in[9];
    (void)n_in; (void)out_size; (void)ws_size;

    const int  B = in_sizes[0];
    const int  N = in_sizes[1] / DA;       // 50000 (multiple of 16)
    const long E = (long)in_sizes[2] / 6;  // 500000
    const int  R = in_sizes[3] / DA;       // 401

    float* out = (float*)d_out;

    // Workspace layout: sproj | agg | rproj | qproj  (~51.5 MB total)
    float* sproj = (float*)d_ws;
    float* agg   = sproj + (size_t)N * DA;
    float* rproj = agg   + (size_t)N * DA;
    float* qproj = rproj + (size_t)R * DA;

    // 1) zero agg
    zero_f32<<<2048, 256, 0, stream>>>(agg, (long)N * DA);

    // 2) rproj / qproj (tiny)
    small_proj<<<R + B, DA, 0, stream>>>(rela, Wr, q_rel, Wqr_w, Wqr_b,
                                         rproj, qproj, R);

    // 3) sproj = hidden @ Ws   (fp32 WMMA, W staged in LDS, strip per wave)
    const int strips = N / 16;                         // 3125
    const int blocks = (strips + 7) / 8;               // 8 strips / block
    gemm128_wmma<<<blocks, 256, 0, stream>>>(hidden, Ws, sproj, N, 0);

    // 4) fused edge phase: gather + relu + dot + sigmoid + atomic scatter
    edge_msg<<<2048, 256, (size_t)B * DA * sizeof(float), stream>>>(
        edges, sproj, rproj, qproj, hidden, rela, Wa, agg, E, B);

    // 5) out = relu(agg @ Wh)  (fp32 WMMA)
    gemm128_wmma<<<blocks, 256, 0, stream>>>(agg, Wh, out, N, 1);
}